// HausdorffERLoss_36550171689530
// MI455X (gfx1250) — compile-verified
//
#include <hip/hip_runtime.h>
#include <stdint.h>

#define HDIM 1024
#define WDIM 1024
#define NB   32
#define IMG_PX (HDIM * WDIM)
#define ER_BYTES ((size_t)NB * IMG_PX * 4)

#define TILE_R 32
#define TILE_C 256
#define BANDS  (WDIM / TILE_C)             // 4
#define STRIPS (HDIM / TILE_R)             // 32
#define BLOCKS_PER_IMG (BANDS * STRIPS)    // 128
#define NTHREADS 256
#define EROSIONS 10

#define USE_TDM 1

typedef uint32_t u32x4 __attribute__((ext_vector_type(4)));
typedef uint32_t u32x8 __attribute__((ext_vector_type(8)));

// ---- CDNA5 Tensor Data Mover: one-descriptor 2D strided tile -> LDS ---------
// D# packing per CDNA5 ISA ch.8 (group0 128b, group1 256b; groups 2/3 NULL).
__device__ __forceinline__ void tdm_load_tile_2d(uint32_t lds_addr, const void* gaddr,
                                                 uint32_t tensor_d0, uint32_t tensor_d1,
                                                 uint32_t tile_d0, uint32_t tile_d1,
                                                 uint32_t stride0_elems) {
    uint64_t ga = (uint64_t)(uintptr_t)gaddr;
    u32x4 g0;
    g0.x = 1u;                                   // [1:0] count=1 (valid user D#)
    g0.y = lds_addr;                             // [63:32] LDS byte address
    g0.z = (uint32_t)ga;                         // [95:64] global addr lo
    g0.w = (uint32_t)(ga >> 32) | (2u << 30);    // [120:96] addr hi | type=2 (image)
    u32x8 g1;
    g1.s0 = 2u << 16;                            // workgroup_mask=0, data_size=2 (4B)
    g1.s1 = (tensor_d0 & 0xFFFFu) << 16;         // tensor_dim0[15:0] (bits 63:48)
    g1.s2 = (tensor_d0 >> 16) | ((tensor_d1 & 0xFFFFu) << 16);   // dim0 hi | dim1 lo
    g1.s3 = (tensor_d1 >> 16) | (tile_d0 << 16); // dim1 hi | tile_dim0
    g1.s4 = tile_d1;                             // tile_dim1 | (tile_dim2=0)<<16
    g1.s5 = stride0_elems;                       // tensor_dim0_stride[31:0]
    g1.s6 = 0u;                                  // stride0 hi | stride1 lo
    g1.s7 = 0u;                                  // stride1 hi
    asm volatile("tensor_load_to_lds %0, %1" :: "s"(g0), "s"(g1) : "memory");
}
__device__ __forceinline__ void wait_tensor0() {
#if __has_builtin(__builtin_amdgcn_s_wait_tensorcnt)
    __builtin_amdgcn_s_wait_tensorcnt(0);
#else
    asm volatile("s_wait_tensorcnt 0x0" ::: "memory");
#endif
}

// ---- fallback: CDNA5 async direct-to-LDS per-lane loads (ASYNCcnt) ----------
__device__ __forceinline__ void async_load_b128(uint32_t lds_byte_addr, const void* gaddr) {
    asm volatile("global_load_async_to_lds_b128 %0, %1, off"
                 :: "v"(lds_byte_addr), "v"(gaddr) : "memory");
}
__device__ __forceinline__ void wait_async0() {
#if __has_builtin(__builtin_amdgcn_s_wait_asynccnt)
    __builtin_amdgcn_s_wait_asynccnt(0);
#else
    asm volatile("s_wait_asynccnt 0x0" ::: "memory");
#endif
}

// ---- bound = (pred - target)^2, vectorized b128 -----------------------------
__global__ void bound_kernel(const float* __restrict__ pred,
                             const float* __restrict__ targ,
                             float* __restrict__ er) {
    size_t i = ((size_t)blockIdx.x * NTHREADS + threadIdx.x) * 4;
    float4 p = *(const float4*)(pred + i);
    float4 t = *(const float4*)(targ + i);
    float4 o;
    o.x = (p.x - t.x) * (p.x - t.x);
    o.y = (p.y - t.y) * (p.y - t.y);
    o.z = (p.z - t.z) * (p.z - t.z);
    o.w = (p.w - t.w) * (p.w - t.w);
    *(float4*)(er + i) = o;
}

// ---- fused: normalized 3x3-cross conv -> relu(x-0.5) + per-image stats ------
__global__ __launch_bounds__(NTHREADS)
void conv_kernel(const float* __restrict__ src, float* __restrict__ dst,
                 const float* __restrict__ scaleArr, const float* __restrict__ offArr,
                 unsigned* __restrict__ umin, unsigned* __restrict__ umax,
                 float* __restrict__ partial) {
    __shared__ __align__(16) float tile[TILE_R + 2][TILE_C];
    __shared__ float redS[NTHREADS], redMn[NTHREADS], redMx[NTHREADS];

    const int img   = blockIdx.y;
    const int band  = blockIdx.x % BANDS;
    const int strip = blockIdx.x / BANDS;
    const int c0 = band * TILE_C;
    const int r0 = strip * TILE_R;
    const int t  = threadIdx.x;
    const float* im = src + (size_t)img * IMG_PX;

    const float sc    = scaleArr[img];
    const float scoff = sc * offArr[img];

#if USE_TDM
    // Tile rows r0-1 .. r0+TILE_R via one TDM descriptor. Bottom overrun is
    // zero-filled by TDM OOB (tensor_dim1 = rows remaining); top halo (row -1
    // before descriptor origin) is zero-filled manually when r0 == 0.
    const int startRow = (r0 == 0) ? 0 : (r0 - 1);
    const int ldsRow0  = (r0 == 0) ? 1 : 0;
    if (ldsRow0 && t < TILE_C / 4) {
        float4 z = {0.f, 0.f, 0.f, 0.f};
        *(float4*)&tile[0][t * 4] = z;
    }
    if (t < 32) {   // one wave issues the DMA; EXEC ignored by tensor ops
        tdm_load_tile_2d((uint32_t)(uintptr_t)&tile[ldsRow0][0],
                         (const void*)(im + (size_t)startRow * WDIM + c0),
                         (uint32_t)(WDIM - c0),          // cols remaining
                         (uint32_t)(HDIM - startRow),    // rows remaining (OOB -> 0)
                         (uint32_t)TILE_C,
                         (uint32_t)(TILE_R + 2 - ldsRow0),
                         (uint32_t)WDIM);
        wait_tensor0();
    }
    __syncthreads();
#else
    const int nChunks = (TILE_R + 2) * (TILE_C / 4);   // 34*64 = 2176 float4s
    for (int chunk = t; chunk < nChunks; chunk += NTHREADS) {
        const int row = chunk >> 6;
        const int q   = (chunk & 63) * 4;
        const int gr  = r0 - 1 + row;
        float* ldsp = &tile[row][q];
        if (gr >= 0 && gr < HDIM) {
            async_load_b128((uint32_t)(uintptr_t)ldsp,
                            (const void*)(im + (size_t)gr * WDIM + c0 + q));
        } else {
            float4 z = {0.f, 0.f, 0.f, 0.f};
            *(float4*)ldsp = z;
        }
    }
    wait_async0();
    __syncthreads();
#endif

    float lsum = 0.f, lmin = __builtin_inff(), lmax = 0.f;
    const int c = c0 + t;
    float* dstim = dst + (size_t)img * IMG_PX;

    for (int lr = 1; lr <= TILE_R; ++lr) {
        const int r = r0 + lr - 1;
        float Cv = tile[lr][t];
        float Nv = tile[lr - 1][t];
        float Sv = tile[lr + 1][t];
        float Wv, Ev;
        if (t > 0)          Wv = tile[lr][t - 1];
        else                Wv = (c > 0) ? im[(size_t)r * WDIM + c - 1] : 0.f;
        if (t < TILE_C - 1) Ev = tile[lr][t + 1];
        else                Ev = (c < WDIM - 1) ? im[(size_t)r * WDIM + c + 1] : 0.f;

        float dil  = 0.2f * (Cv + Nv + Sv + Wv + Ev);      // conv of RAW field
        int   taps = 1 + (r > 0) + (r < HDIM - 1) + (c > 0) + (c < WDIM - 1);
        float wsum = 0.2f * (float)taps;                   // in-bounds kernel mass
        // conv(normalized) = sc*conv(raw) - sc*off*wsum  (zero padding!)
        float e = fmaxf(sc * dil - scoff * wsum - 0.5f, 0.f);

        dstim[(size_t)r * WDIM + c] = e;
        lsum += e;
        lmin = fminf(lmin, e);
        lmax = fmaxf(lmax, e);
    }

    redS[t] = lsum; redMn[t] = lmin; redMx[t] = lmax;
    __syncthreads();
    for (int s = NTHREADS / 2; s > 0; s >>= 1) {
        if (t < s) {
            redS[t] += redS[t + s];
            redMn[t] = fminf(redMn[t], redMn[t + s]);
            redMx[t] = fmaxf(redMx[t], redMx[t + s]);
        }
        __syncthreads();
    }
    if (t == 0) {
        partial[img * BLOCKS_PER_IMG + blockIdx.x] = redS[0];
        // er >= 0, so uint ordering == float ordering (deterministic atomics)
        atomicMin(&umin[img], __float_as_uint(redMn[0]));
        atomicMax(&umax[img], __float_as_uint(redMx[0]));
    }
}

// ---- per-image: reduce partials, derive (scale,offset), accumulate scalar ---
__global__ void finalize_kernel(unsigned* __restrict__ umin, unsigned* __restrict__ umax,
                                const float* __restrict__ partial,
                                float* __restrict__ scaleArr, float* __restrict__ offArr,
                                double* __restrict__ contrib, int k) {
    __shared__ double s[BLOCKS_PER_IMG];
    const int b = blockIdx.x, t = threadIdx.x;
    s[t] = (double)partial[b * BLOCKS_PER_IMG + t];
    __syncthreads();
    for (int st = BLOCKS_PER_IMG / 2; st > 0; st >>= 1) {
        if (t < st) s[t] += s[t + st];
        __syncthreads();
    }
    if (t == 0) {
        float mn = __uint_as_float(umin[b]);
        float mx = __uint_as_float(umax[b]);
        float ptp = mx - mn;
        float sc, off;
        if (ptp > 0.f) { sc = 1.f / ptp; off = mn; } else { sc = 1.f; off = 0.f; }
        scaleArr[b] = sc;
        offArr[b]   = off;
        double wk = (double)(k + 1) * (double)(k + 1);     // (k+1)^ALPHA, ALPHA=2
        contrib[b] += wk * (s[0] - (double)IMG_PX * (double)off) * (double)sc;
        umin[b] = 0x7F800000u;   // +inf bits
        umax[b] = 0u;            // 0.0f bits (field is non-negative)
    }
}

__global__ void init_kernel(unsigned* umin, unsigned* umax,
                            float* scaleArr, float* offArr, double* contrib) {
    int t = threadIdx.x;
    if (t < NB) {
        umin[t] = 0x7F800000u; umax[t] = 0u;
        scaleArr[t] = 1.f; offArr[t] = 0.f;
        contrib[t] = 0.0;
    }
}

__global__ void epilogue_kernel(const double* __restrict__ contrib, float* __restrict__ out) {
    if (threadIdx.x == 0) {
        double tot = 0.0;
        for (int b = 0; b < NB; ++b) tot += contrib[b];
        out[0] = (float)(tot / ((double)NB * (double)IMG_PX));
    }
}

extern "C" void kernel_launch(void* const* d_in, const int* in_sizes, int n_in,
                              void* d_out, int out_size, void* d_ws, size_t ws_size,
                              hipStream_t stream) {
    (void)in_sizes; (void)n_in; (void)out_size; (void)ws_size;
    const float* pred = (const float*)d_in[0];
    const float* targ = (const float*)d_in[1];
    float* out = (float*)d_out;

    char* ws = (char*)d_ws;
    float* erA = (float*)ws;
    float* erB = (float*)(ws + ER_BYTES);
    char* sb = ws + 2 * ER_BYTES;
    double*   contrib  = (double*)sb;              // 32 * 8
    unsigned* umin     = (unsigned*)(sb + 256);    // 32 * 4
    unsigned* umax     = (unsigned*)(sb + 384);
    float*    scaleArr = (float*)(sb + 512);
    float*    offArr   = (float*)(sb + 640);
    float*    partial  = (float*)(sb + 768);       // 32 * 128 * 4

    init_kernel<<<1, 64, 0, stream>>>(umin, umax, scaleArr, offArr, contrib);

    const int nVec4 = (int)((size_t)NB * IMG_PX / 4);
    bound_kernel<<<nVec4 / NTHREADS, NTHREADS, 0, stream>>>(pred, targ, erA);

    float* src = erA;
    float* dst = erB;
    dim3 grid(BLOCKS_PER_IMG, NB);
    for (int k = 0; k < EROSIONS; ++k) {
        conv_kernel<<<grid, NTHREADS, 0, stream>>>(src, dst, scaleArr, offArr,
                                                   umin, umax, partial);
        finalize_kernel<<<NB, BLOCKS_PER_IMG, 0, stream>>>(umin, umax, partial,
                                                           scaleArr, offArr, contrib, k);
        float* tmp = src; src = dst; dst = tmp;
    }

    epilogue_kernel<<<1, 32, 0, stream>>>(contrib, out);
}